// LinearPotential_86062554677254
// MI455X (gfx1250) — compile-verified
//
#include <hip/hip_runtime.h>

typedef __attribute__((ext_vector_type(2))) float v2f;
typedef __attribute__((ext_vector_type(4))) float v4f;
typedef __attribute__((ext_vector_type(8))) float v8f;

#define N_POINTS 131072
#define N_ANCHORS 2048
#define LOG2E 1.4426950408889634f

__global__ __launch_bounds__(256) void LinearPotential_kernel(
    const float* __restrict__ loc,    // [N,3]
    const float* __restrict__ aloc,   // [M,3]
    const float* __restrict__ acoef,  // [M]
    const float* __restrict__ aparm,  // [M]
    float* __restrict__ out)          // [N]
{
    // Per-anchor transformed data staged in LDS once per block (48 KB).
    __shared__ v4f   sB[N_ANCHORS];   // (2s*ax, 2s*ay, 2s*az, -s)
    __shared__ float sD[N_ANCHORS];   // -s*|a|^2
    __shared__ float sC[N_ANCHORS];   // coeff

    const int tid = threadIdx.x;

    // ---- Phase 1: cooperative anchor transform ----
    for (int j = tid; j < N_ANCHORS; j += 256) {
        float ax = aloc[3 * j + 0];
        float ay = aloc[3 * j + 1];
        float az = aloc[3 * j + 2];
        float w  = aparm[j];
        float s  = (0.5f * LOG2E) / (w * w);   // exponent pre-scaled for exp2
        v4f b;
        b.x = 2.0f * s * ax;
        b.y = 2.0f * s * ay;
        b.z = 2.0f * s * az;
        b.w = -s;
        sB[j] = b;
        sD[j] = -s * (ax * ax + ay * ay + az * az);
        sC[j] = acoef[j];
    }
    __syncthreads();

    // ---- Phase 2: WMMA over 16 points x 2048 anchors per wave ----
    const int lane     = tid & 31;
    const int waveId   = tid >> 5;
    const int tileBase = blockIdx.x * 128 + waveId * 16;
    const int pt       = tileBase + (lane & 15);   // same point in both halves

    const float x = loc[3 * pt + 0];
    const float y = loc[3 * pt + 1];
    const float z = loc[3 * pt + 2];
    const float xsq = x * x + y * y + z * z;

    const bool hiHalf = (lane >= 16);
    // A (16x4 f32): lanes 0-15 -> K=0,1 ; lanes 16-31 -> K=2,3
    v2f A;
    A.x = hiHalf ? z   : x;
    A.y = hiHalf ? xsq : y;

    float acc[8] = {0.f, 0.f, 0.f, 0.f, 0.f, 0.f, 0.f, 0.f};
    const v8f czero = {};
    const float* pB = (const float*)sB;
    const int sel = hiHalf ? 2 : 0;

    for (int chunk = 0; chunk < N_ANCHORS / 16; ++chunk) {
        const int j = chunk * 16 + (lane & 15);
        // B (4x16 f32): lanes 0-15 -> rows K=0,1 ; lanes 16-31 -> rows K=2,3
        v2f B;
        B.x = pB[4 * j + sel + 0];
        B.y = pB[4 * j + sel + 1];
        const float d = sD[j];   // per-column (anchor) exponent offset
        const float c = sC[j];   // per-column coefficient

        // t[M][N] = 2s x.a - s|x|^2  (C = inline 0 -> no input dependency,
        // WMMA issues immediately after the B-tile ds_loads land)
        v8f t = __builtin_amdgcn_wmma_f32_16x16x4_f32(
            /*neg_a=*/false, A, /*neg_b=*/false, B,
            /*c_mod=*/(short)0, czero, /*reuse_a=*/false, /*reuse_b=*/false);

#pragma unroll
        for (int r = 0; r < 8; ++r) {
            float p = __builtin_amdgcn_exp2f(t[r] + d);  // v_add (dual) + v_exp_f32 (TRANS)
            acc[r] = __builtin_fmaf(c, p, acc[r]);       // v_fmac_f32 (dual)
        }
    }

    // ---- Reduce across the 16 anchors (lanes) in each half, then store ----
#pragma unroll
    for (int r = 0; r < 8; ++r) {
        float v = acc[r];
#pragma unroll
        for (int m = 1; m < 16; m <<= 1)
            v += __shfl_xor(v, m, 32);
        // D layout: lane<16 -> point tileBase+r ; lane>=16 -> point tileBase+r+8
        if ((lane & 15) == 0)
            out[tileBase + r + (lane >> 4) * 8] = v;
    }
}

extern "C" void kernel_launch(void* const* d_in, const int* in_sizes, int n_in,
                              void* d_out, int out_size, void* d_ws, size_t ws_size,
                              hipStream_t stream) {
    (void)in_sizes; (void)n_in; (void)out_size; (void)d_ws; (void)ws_size;
    const float* loc   = (const float*)d_in[0];
    const float* aloc  = (const float*)d_in[1];
    const float* acoef = (const float*)d_in[2];
    const float* aparm = (const float*)d_in[3];
    float* out = (float*)d_out;

    dim3 grid(N_POINTS / 128);   // 8 waves/block, 16 points/wave
    dim3 block(256);
    LinearPotential_kernel<<<grid, block, 0, stream>>>(loc, aloc, acoef, aparm, out);
}